// Global_Attn_3461743640963
// MI455X (gfx1250) — compile-verified
//
#include <hip/hip_runtime.h>

// MI455X (gfx1250) fused global-attention head-mixing kernel, round 3.
//
// Roofline: output = 8*8*1024*1024 fp32 = 256 MB -> HBM-store bound
// (~11 us at 23.3 TB/s). Inputs (~34 MB) stay L2-resident (192 MB L2).
// QK^T and the mask-weight projection run on V_WMMA_F32_16X16X4_F32.
// The mask-weight tile is computed once per (q,k) tile and shared by all 8
// batches; the 256 MB output is streamed with non-temporal stores.
//
// Round-3 change: decouple WMMA accumulators from the epilogue through LDS
// (sAll staging) so no code needs dynamic ext-vector indexing and the
// epilogue runs as compact rolled loops -> VGPR pressure drops from >256 to
// well under 128 (no more s_set_vgpr_msb / SGPR spills).

typedef __attribute__((ext_vector_type(2))) float v2f;
typedef __attribute__((ext_vector_type(4))) float v4f;
typedef __attribute__((ext_vector_type(8))) float v8f;

#define N_TOK   1024
#define D_HEAD  64
#define GH_N    4
#define LH_N    8
#define B_N     8
#define L_MASK  4
// mw rows padded 32 -> 36 floats (144 B): 16 lanes * b128 hit 16 disjoint
// 4-bank groups (n*36 mod 64 distinct, spacing 4) -> conflict-free.
#define MW_STRIDE 36
#define MW_FLOATS (256 * MW_STRIDE)          // 9216
#define SALL_OFF  MW_FLOATS                  // sAll[b][g][pos] after mw
#define LDS_FLOATS (MW_FLOATS + B_N * GH_N * 256)  // 9216 + 8192 = 17408 (68 KB)

__launch_bounds__(256, 1)
__attribute__((amdgpu_waves_per_eu(8)))
__global__ void fused_global_attn_kernel(const float* __restrict__ q,
                                         const float* __restrict__ k,
                                         const float* __restrict__ masks,
                                         const float* __restrict__ mask_proj,
                                         const float* __restrict__ head_w,
                                         const float* __restrict__ head_b,
                                         float* __restrict__ out) {
    __shared__ float lds[LDS_FLOATS];

    const int lane = threadIdx.x & 31;
    const int wave = threadIdx.x >> 5;      // 8 waves; wave == batch
    const int kt   = blockIdx.x & 63;
    const int qt   = blockIdx.x >> 6;
    const int q0   = qt << 4;
    const int k0   = kt << 4;
    const int lm   = lane & 15;             // M/N index within 16
    const int lh   = lane >> 4;             // lane half (K split of f32 A/B layout)

    // ---- Phase 1: mw[pos][g*8+h'] = sum_l masks[q,k,l] * proj[l, g*8+h']
    {
        v2f bp0, bp1;                        // mask_proj 4x16 column halves
        const int kd0 = lh * 2;
        bp0.x = mask_proj[(kd0 + 0) * 32 + lm];
        bp0.y = mask_proj[(kd0 + 1) * 32 + lm];
        bp1.x = mask_proj[(kd0 + 0) * 32 + 16 + lm];
        bp1.y = mask_proj[(kd0 + 1) * 32 + 16 + lm];
        #pragma unroll
        for (int rr = 0; rr < 2; ++rr) {
            const int r = wave * 2 + rr;     // qi row; WMMA M dim = ki index
            const float* mrow = masks
                + (((size_t)(q0 + r)) * N_TOK + (size_t)(k0 + lm)) * L_MASK + lh * 2;
            v2f a;
            a.x = mrow[0];
            a.y = mrow[1];
            v8f c0 = {};
            v8f c1 = {};
            c0 = __builtin_amdgcn_wmma_f32_16x16x4_f32(false, a, false, bp0,
                                                       (short)0, c0, false, false);
            c1 = __builtin_amdgcn_wmma_f32_16x16x4_f32(false, a, false, bp1,
                                                       (short)0, c1, false, false);
            #pragma unroll
            for (int v = 0; v < 8; ++v) {    // D elem (ki = v+8*lh, gh = col+lm)
                const int ki = v + 8 * lh;
                lds[(r * 16 + ki) * MW_STRIDE + lm]      = c0[v];
                lds[(r * 16 + ki) * MW_STRIDE + 16 + lm] = c1[v];
            }
        }
    }

    // ---- Phase 2a: S_g = Q K^T for this wave's batch, staged straight to LDS.
    {
        const int b = wave;
        #pragma unroll
        for (int g = 0; g < GH_N; ++g) {
            // A[m][kd]=q[b,g,q0+m,d0+kd]; B[kd][n]=k[b,g,k0+n,d0+kd]; both are
            // per-lane pattern token=lane&15, d=d0+(lane>>4)*2 -> float2 loads.
            const float* qbase = q
                + ((((size_t)b * GH_N + g) * N_TOK) + (size_t)(q0 + lm)) * D_HEAD + lh * 2;
            const float* kbase = k
                + ((((size_t)b * GH_N + g) * N_TOK) + (size_t)(k0 + lm)) * D_HEAD + lh * 2;
            v8f c = {};
            #pragma unroll
            for (int d0 = 0; d0 < D_HEAD; d0 += 4) {
                v2f av = *(const v2f*)(qbase + d0);
                v2f bv = *(const v2f*)(kbase + d0);
                c = __builtin_amdgcn_wmma_f32_16x16x4_f32(false, av, false, bv,
                                                          (short)0, c, false, false);
            }
            // sAll[b][g][pos], pos = (v+8*lh)*16 + lm. Single base reg +
            // static 16-bit immediates; accumulator dies here.
            float* sb = &lds[SALL_OFF + (b * GH_N + g) * 256 + lh * 128 + lm];
            #pragma unroll
            for (int v = 0; v < 8; ++v)
                sb[v * 16] = c[v];
        }
    }
    __syncthreads();

    // ---- Phase 2b: per-position epilogue, rolled over batches.
    {
        const int pos = threadIdx.x;         // 0..255: qi = pos>>4, ki = pos&15
        const int qi  = pos >> 4;
        const int ki  = pos & 15;

        // Hoist this position's 32 mask-weights once (8 x b128, conflict-free).
        v4f w[8];
        #pragma unroll
        for (int i = 0; i < 8; ++i)
            w[i] = *(const v4f*)&lds[pos * MW_STRIDE + i * 4];

        const float* sp = &lds[SALL_OFF + pos];
        float* obase = out + (size_t)(q0 + qi) * N_TOK + (size_t)(k0 + ki);

        #pragma unroll 1
        for (int b = 0; b < B_N; ++b) {
            // 4 ds_load_b32 off one base with immediate offsets.
            float s0 = sp[(b * GH_N + 0) * 256];
            float s1 = sp[(b * GH_N + 1) * 256];
            float s2 = sp[(b * GH_N + 2) * 256];
            float s3 = sp[(b * GH_N + 3) * 256];

            float t[LH_N];
            #pragma unroll
            for (int hp = 0; hp < LH_N; ++hp) {
                float acc;
                acc = s0 * w[hp >> 2][hp & 3];                 // g=0: w[0+hp]
                acc = fmaf(s1, w[(8  + hp) >> 2][hp & 3], acc); // g=1
                acc = fmaf(s2, w[(16 + hp) >> 2][hp & 3], acc); // g=2
                acc = fmaf(s3, w[(24 + hp) >> 2][hp & 3], acc); // g=3
                t[hp] = fmaxf(acc, 0.0f);                       // relu
            }

            // out[b,h,q,k]; head_w/head_b read uniformly -> scalar loads.
            float* ob = obase + (size_t)b * LH_N * N_TOK * N_TOK;
            #pragma unroll
            for (int h = 0; h < LH_N; ++h) {
                float acc = head_b[h];
                #pragma unroll
                for (int hp = 0; hp < LH_N; ++hp)
                    acc = fmaf(t[hp], head_w[h * LH_N + hp], acc);
                // Write-once 256 MB stream: non-temporal, keep L2 for inputs.
                __builtin_nontemporal_store(acc, ob + (size_t)h * N_TOK * N_TOK);
            }
        }
    }
}

extern "C" void kernel_launch(void* const* d_in, const int* in_sizes, int n_in,
                              void* d_out, int out_size, void* d_ws, size_t ws_size,
                              hipStream_t stream) {
    (void)in_sizes; (void)n_in; (void)out_size; (void)d_ws; (void)ws_size;
    const float* q         = (const float*)d_in[0];
    const float* k         = (const float*)d_in[1];
    const float* masks     = (const float*)d_in[2];
    const float* mask_proj = (const float*)d_in[3];
    const float* head_w    = (const float*)d_in[4];
    const float* head_b    = (const float*)d_in[5];
    float* out = (float*)d_out;

    // One block per 16x16 (q,k) tile; 8 waves = 8 batches per block so the
    // mask-weight tile is computed once and shared.
    const int tiles = (N_TOK / 16) * (N_TOK / 16);  // 4096
    fused_global_attn_kernel<<<tiles, 256, 0, stream>>>(q, k, masks, mask_proj,
                                                        head_w, head_b, out);
}